// TaggerGreedyDecoder_46084999086670
// MI455X (gfx1250) — compile-verified
//
#include <hip/hip_runtime.h>
#include <stdint.h>

// Problem constants (from reference: B=128, T=1024, N=128, GO=1, EOS=2)
#define B_   128
#define T_   1024
#define N_   128
#define GO_  1
#define EOS_ 2
#define NEGV (-10000.0f)
#define NPAD 132   // padded LDS row stride (floats): +4 DWORDs -> 16B-aligned rows,
                   // conflict-free ds_load_b128 (lanes 0..15 cover all 64 banks/phase)

// ---- LDS layout (bytes) ----
#define OFF_TRANS   0                           // 128*132*4 = 67584
#define OFF_ALPHAS  (OFF_TRANS  + N_*NPAD*4)    // 512 (16B aligned)
#define OFF_PBEST   (OFF_ALPHAS + N_*4)         // 512
#define OFF_PIDX    (OFF_PBEST  + N_*4)         // 512
#define OFF_LSEM    (OFF_PIDX   + N_*4)         // 16
#define OFF_LSES    (OFF_LSEM   + 4*4)          // 16
#define OFF_PATH    (OFF_LSES   + 4*4)          // 1024 (u8 path)
#define OFF_BP      (OFF_PATH   + T_)           // 131072 (u8 backpointers)
#define SMEM_BYTES  (OFF_BP + T_*N_)            // = 201248 < 320 KB WGP LDS

#if __has_builtin(__builtin_amdgcn_tensor_load_to_lds)
#define USE_TDM 1
#endif

typedef __attribute__((ext_vector_type(4))) unsigned int u32x4;
typedef __attribute__((ext_vector_type(8))) int          i32x8;
typedef __attribute__((ext_vector_type(4))) int          i32x4;

__launch_bounds__(256, 1)
__global__ void viterbi_tagger_kernel(const float* __restrict__ unaries,   // (B,T,N)
                                      const float* __restrict__ trans,     // (N,N)
                                      const int*   __restrict__ lengths,   // (B,)
                                      float* __restrict__ out_path,        // (B,T) as float
                                      float* __restrict__ out_score)       // (B,)
{
    extern __shared__ char smem[];
    float*         lds_trans  = (float*)(smem + OFF_TRANS);
    float*         lds_alphas = (float*)(smem + OFF_ALPHAS);
    float*         lds_pbest  = (float*)(smem + OFF_PBEST);
    int*           lds_pidx   = (int*)  (smem + OFF_PIDX);
    float*         lds_lsem   = (float*)(smem + OFF_LSEM);
    float*         lds_lses   = (float*)(smem + OFF_LSES);
    unsigned char* lds_path   = (unsigned char*)(smem + OFF_PATH);
    unsigned char* lds_bp     = (unsigned char*)(smem + OFF_BP);

    const int b    = blockIdx.x;
    const int tid  = threadIdx.x;
    const int i    = tid & (N_ - 1);   // output tag this thread owns
    const int h    = tid >> 7;         // column half: 0 -> j in [0,64), 1 -> [64,128)
    const int wav  = tid >> 5;         // wave32 index 0..7
    const int lane = tid & 31;
    const int len  = lengths[b];
    const float* __restrict__ ub = unaries + (size_t)b * T_ * N_;

    // ---- Stage transition matrix into LDS with +4-DWORD row padding (stride 132) ----
#ifdef USE_TDM
    if (wav == 0) {   // wave-uniform guard: TDM issues once (EXEC ignored by TDM)
        uint32_t lds_base = (uint32_t)(uintptr_t)(void*)smem;  // flat low 32 bits == LDS offset
        uint64_t ga = (uint64_t)(uintptr_t)trans;
        u32x4 g0;
        g0.x = 1u;                                   // count = 1 (valid user descriptor)
        g0.y = lds_base + OFF_TRANS;                 // lds_addr
        g0.z = (uint32_t)ga;                         // global_addr[31:0]
        g0.w = (uint32_t)((ga >> 32) & 0x1FFFFFFu)   // global_addr[56:32]
             | (2u << 30);                           // type = 2 ("image")
        i32x8 g1;
        g1[0] = (2 << 16)        // data_size = 4B
              | (1 << 20)        // pad_enable: pad LDS destination
              | (6 << 22)        // pad_interval code 6 -> every 128 DWORDs
              | (3 << 25);       // pad_amount  code 3 -> 4 DWORDs  => row stride 132
        g1[1] = (N_ << 16);      // tensor_dim0[15:0] in bits[63:48]
        g1[2] = (N_ << 16);      // tensor_dim1[15:0] in bits[95:80]
        g1[3] = (N_ << 16);      // tile_dim0 in bits[127:112]
        g1[4] = N_;              // tile_dim1 (tile_dim2 = 0)
        g1[5] = N_;              // tensor_dim0_stride[31:0]
        g1[6] = (int)(((uint32_t)(N_ * N_) & 0xFFFFu) << 16);  // dim1_stride[15:0]
        g1[7] = (N_ * N_) >> 16;                               // dim1_stride[47:16]
        i32x4 gz = {0, 0, 0, 0};
#if defined(__clang_major__) && (__clang_major__ >= 23)
        i32x8 gz8 = {0, 0, 0, 0, 0, 0, 0, 0};
        __builtin_amdgcn_tensor_load_to_lds(g0, g1, gz, gz, gz8, 0);
#else
        __builtin_amdgcn_tensor_load_to_lds(g0, g1, gz, gz, 0);
#endif
#if __has_builtin(__builtin_amdgcn_s_wait_tensorcnt)
        __builtin_amdgcn_s_wait_tensorcnt(0);
#else
        asm volatile("s_wait_tensorcnt 0x0" ::: "memory");
#endif
    }
#else
    for (int k = tid; k < N_ * N_; k += 256)
        lds_trans[(k >> 7) * NPAD + (k & (N_ - 1))] = trans[k];
#endif

    // init alphas = log_softmax(one-hot GO, rest NEG) == {0 at GO, NEG elsewhere} in f32
    if (h == 0) lds_alphas[i] = (i == GO_) ? 0.0f : NEGV;
    __syncthreads();

    // ---- forward scan over T (sequential; batch-parallel across WGs) ----
    float u = 0.0f;
    if (h == 0) u = ub[i];                 // unary row for t = 0

    for (int t = 0; t < T_; ++t) {
        // per-wave partial stats for log-sum-exp of current unary row (waves 0..3)
        if (h == 0) {
            float m = u;
            for (int off = 16; off > 0; off >>= 1)
                m = fmaxf(m, __shfl_xor(m, off, 32));
            float e = __expf(u - m);
            for (int off = 16; off > 0; off >>= 1)
                e += __shfl_xor(e, off, 32);
            if (lane == 0) { lds_lsem[wav] = m; lds_lses[wav] = e; }
        }

        // tropical inner product over this thread's 64-column slice, via ds_load_b128
        const float4* __restrict__ tr4 = (const float4*)(lds_trans + i * NPAD + (h << 6));
        const float4* __restrict__ al4 = (const float4*)(lds_alphas + (h << 6));
        float best = -1e30f; int bidx = 0;
        #pragma unroll
        for (int q = 0; q < 16; ++q) {
            float4 av = al4[q];            // broadcast (all lanes same address)
            float4 tv = tr4[q];            // conflict-free b128 (stride-132 rows)
            float s;
            s = av.x + tv.x; if (s > best) { best = s; bidx = 4 * q + 0; }
            s = av.y + tv.y; if (s > best) { best = s; bidx = 4 * q + 1; }
            s = av.z + tv.z; if (s > best) { best = s; bidx = 4 * q + 2; }
            s = av.w + tv.w; if (s > best) { best = s; bidx = 4 * q + 3; }
        }
        bidx += (h << 6);                  // strict '>' => first-occurrence argmax

        if (h == 1) { lds_pbest[i] = best; lds_pidx[i] = bidx; }
        __syncthreads();

        if (h == 0) {
            float ob = lds_pbest[i];
            if (ob > best) { best = ob; bidx = lds_pidx[i]; }   // tie -> lower half wins
            // combine 4 wave stats -> logsumexp
            float m0 = lds_lsem[0], m1 = lds_lsem[1], m2 = lds_lsem[2], m3 = lds_lsem[3];
            float gm = fmaxf(fmaxf(m0, m1), fmaxf(m2, m3));
            float ss = lds_lses[0] * __expf(m0 - gm) + lds_lses[1] * __expf(m1 - gm)
                     + lds_lses[2] * __expf(m2 - gm) + lds_lses[3] * __expf(m3 - gm);
            float lse = gm + __logf(ss);

            float na = best + (u - lse);
            if (t < len) lds_alphas[i] = na;                 // length-masked update
            lds_bp[t * N_ + i] = (unsigned char)bidx;        // backpointer always recorded

            // software pipeline: fetch next unary row; latency hides behind inner loop
            if (t + 1 < T_) u = ub[(t + 1) * N_ + i];
            if (t + 8 < T_) __builtin_prefetch(&ub[(t + 8) * N_ + i], 0, 1); // global_prefetch_b8
        }
        __syncthreads();
    }

    // ---- terminal scores, argmax, backtrace ----
    if (h == 0) lds_pbest[i] = lds_alphas[i] + lds_trans[EOS_ * NPAD + i];
    __syncthreads();

    if (tid == 0) {
        float bs = lds_pbest[0]; int bt = 0;
        for (int k = 1; k < N_; ++k) {
            float v = lds_pbest[k];
            if (v > bs) { bs = v; bt = k; }   // first-occurrence argmax
        }
        out_score[b] = bs;
        int cur = bt;
        const int rev_len = T_ - len - 1;
        for (int ii = 0; ii < T_; ++ii) {     // ii indexes reversed scan
            int t = T_ - 1 - ii;
            lds_path[t] = (unsigned char)cur; // record BEFORE update (matches reference)
            if (ii > rev_len) cur = lds_bp[t * N_ + cur];
        }
    }
    __syncthreads();

    // coalesced path writeback
    for (int t = tid; t < T_; t += 256)
        out_path[(size_t)b * T_ + t] = (float)lds_path[t];
}

extern "C" void kernel_launch(void* const* d_in, const int* in_sizes, int n_in,
                              void* d_out, int out_size, void* d_ws, size_t ws_size,
                              hipStream_t stream) {
    const float* unaries = (const float*)d_in[0];   // (B,T,N) f32
    const float* trans   = (const float*)d_in[1];   // (N,N)  f32
    const int*   lengths = (const int*)  d_in[2];   // (B,)   i32

    float* out_path  = (float*)d_out;                       // B*T elements
    float* out_score = (float*)d_out + (size_t)B_ * T_;     // B elements

    // 201,248 B dynamic LDS > default cap: raise the attribute (deterministic, capture-safe)
    (void)hipFuncSetAttribute((const void*)viterbi_tagger_kernel,
                              hipFuncAttributeMaxDynamicSharedMemorySize, SMEM_BYTES);

    viterbi_tagger_kernel<<<B_, 256, SMEM_BYTES, stream>>>(unaries, trans, lengths,
                                                           out_path, out_score);
}